// My_model_58918361366790
// MI455X (gfx1250) — compile-verified
//
#include <hip/hip_runtime.h>

typedef __attribute__((ext_vector_type(2))) float v2f;
typedef __attribute__((ext_vector_type(8))) float v8f;
typedef __attribute__((ext_vector_type(4))) unsigned int v4u;
typedef __attribute__((ext_vector_type(4))) int v4i;
typedef __attribute__((ext_vector_type(8))) int v8i;

#define N_WORDS 50000
#define LMAX 12
#define D 512
#define VOCAB 1552
#define CVOCAB 80
#define QSTRIDE 81            // pad to kill LDS bank conflicts on A-frag loads
#define WAVES 8
#define NTILES (N_WORDS / 16) // 3125 exactly

// ---------------------------------------------------------------------------
// Kernel 0: zero the 512-float word_weights accumulator in workspace
// ---------------------------------------------------------------------------
__global__ void zero_ws(float* ws) {
  int i = blockIdx.x * blockDim.x + threadIdx.x;
  if (i < D) ws[i] = 0.f;
}

// ---------------------------------------------------------------------------
// Kernel 1: fused embed + char attention (as Q@T GEMM via WMMA) + sigmoid
//           + word swish weights + word_weights accumulation.
//           Char table staged to LDS via TDM (tensor_load_to_lds).
// ---------------------------------------------------------------------------
__global__ __launch_bounds__(256) void fused_kernel(
    const int* __restrict__ word_ids,
    const int* __restrict__ char_ids,
    const int* __restrict__ char_lens,
    const float* __restrict__ word_embed,
    const float* __restrict__ char_embed,
    const float* __restrict__ char_proj_w,
    const float* __restrict__ char_proj_b,
    const float* __restrict__ word_proj_w,
    const float* __restrict__ word_proj_b,
    float* __restrict__ out_ae,
    float* __restrict__ ww_accum)
{
  __shared__ float sT[CVOCAB * D];            // 160 KB char table (B matrix)
  __shared__ float sQ[WAVES * 16 * QSTRIDE];  // per-wave 16x80 Q tiles
  __shared__ float sS[CVOCAB];                // per-char-id score s[v]
  __shared__ float sCW[D];
  __shared__ float sWPW[D];
  __shared__ float sWW[D];                    // block-local word_weights accum

  const int tid  = threadIdx.x;
  const int lane = tid & 31;
  const int wave = tid >> 5;

  // --- TDM: async DMA the whole 80x512 f32 char table into LDS ------------
  // D# group0: count=1 | lds_addr[63:32] | global_addr[120:64] | type=2[127:126]
  // D# group1: data_size=2 (4B) | tensor_dim0=512 | tensor_dim1=80
  //            | tile_dim0=512 | tile_dim1=80 | tensor_dim0_stride=512
  if (wave == 0) {
    const unsigned long long ga = (unsigned long long)(uintptr_t)char_embed;
    const unsigned int lds_off  = (unsigned int)(uintptr_t)(void*)sT;
    v4u g0;
    g0[0] = 1u;                                   // count=1, user descriptor
    g0[1] = lds_off;                              // lds_addr (bytes)
    g0[2] = (unsigned int)(ga & 0xFFFFFFFFu);     // global_addr[31:0]
    g0[3] = (unsigned int)((ga >> 32) & 0x1FFFFFFu) | (2u << 30); // [56:32]|type=2
    v8i g1;
    g1[0] = (int)(2u << 16);                      // workgroup_mask=0, data_size=4B
    g1[1] = (int)(512u << 16);                    // tensor_dim0[15:0] @ bits 63:48
    g1[2] = (int)(80u << 16);                     // dim0 hi=0 | tensor_dim1[15:0]
    g1[3] = (int)(512u << 16);                    // dim1 hi=0 | tile_dim0=512
    g1[4] = 80;                                   // tile_dim1=80, tile_dim2=0
    g1[5] = 512;                                  // tensor_dim0_stride lo
    g1[6] = 0;                                    // stride hi | dim1_stride lo
    g1[7] = 0;
    v4i gz = {0, 0, 0, 0};
#if defined(__clang_major__) && (__clang_major__ >= 23)
    v8i gz8 = {0, 0, 0, 0, 0, 0, 0, 0};
    __builtin_amdgcn_tensor_load_to_lds(g0, g1, gz, gz, gz8, 0);
#else
    __builtin_amdgcn_tensor_load_to_lds(g0, g1, gz, gz, 0);
#endif
  }

  // overlap: small projection vectors via normal loads while the DMA runs
  for (int i = tid; i < D; i += 256) {
    sCW[i]  = char_proj_w[i];
    sWPW[i] = word_proj_w[i];
    sWW[i]  = 0.f;
  }

  if (wave == 0) __builtin_amdgcn_s_wait_tensorcnt(0);
  __syncthreads();   // publishes TDM-written LDS to all waves

  // --- s[v] = dot(T[v,:], char_proj_w) + b  (one wave per 10 rows) ---
  {
    const float cb = char_proj_b[0];
    for (int v = wave; v < CVOCAB; v += WAVES) {
      float p = 0.f;
      for (int d = lane; d < D; d += 32) p += sT[v * D + d] * sCW[d];
      for (int off = 16; off > 0; off >>= 1) p += __shfl_xor(p, off, 32);
      if (lane == 0) sS[v] = p + cb;
    }
  }
  __syncthreads();

  const int tileId = blockIdx.x * WAVES + wave;   // one 16-word tile per wave
  if (tileId < NTILES) {
    const int w0 = tileId * 16;
    float* q = &sQ[wave * 16 * QSTRIDE];

    // --- build Q[16,80]: q[n,v] = s[v] * count(char v in word n) ---
    for (int i = lane; i < 16 * QSTRIDE; i += 32) q[i] = 0.f;
    if (lane < 16) {
      const int w   = w0 + lane;
      const int len = char_lens[w];
      for (int l = 0; l < LMAX; ++l) {
        if (l < len) {
          const int c = char_ids[w * LMAX + l];
          q[lane * QSTRIDE + c] += sS[c];     // lane owns its row; in-order LDS
        }
      }
    }

    // --- load A fragments once: 20 chunks of 16x4 f32 ---
    // A layout (16x4 f32): lanes 0-15 hold (K0,K1), lanes 16-31 hold (K2,K3)
    const int half2 = (lane >> 4) << 1;
    const int mrow  = lane & 15;
    v2f a[20];
    #pragma unroll
    for (int kc = 0; kc < 20; ++kc) {
      const int kb = kc * 4 + half2;
      a[kc].x = q[mrow * QSTRIDE + kb];
      a[kc].y = q[mrow * QSTRIDE + kb + 1];
    }

    const int   halfw = (lane >> 4) << 3;   // C/D rows: VGPR r -> word r (+8 hi half)
    const float wpb   = word_proj_b[0];
    float wap[8];
    #pragma unroll
    for (int r = 0; r < 8; ++r) wap[r] = 0.f;

    // --- pass 1: char_pool GEMM (WMMA) + fused sigmoid epilogue ---
    for (int dt = 0; dt < 32; ++dt) {
      const int dcol = dt * 16 + mrow;
      v8f acc = {};
      #pragma unroll
      for (int kc = 0; kc < 20; ++kc) {
        const int kb = kc * 4 + half2;
        v2f b;                                 // B = T[k, dcol], mirror layout of A
        b.x = sT[kb * D + dcol];
        b.y = sT[(kb + 1) * D + dcol];
        acc = __builtin_amdgcn_wmma_f32_16x16x4_f32(
            false, a[kc], false, b, (short)0, acc, false, false);
      }
      #pragma unroll
      for (int r = 0; r < 8; ++r) {
        const int   word = w0 + halfw + r;
        const int   wid  = word_ids[word];
        const float we   = word_embed[wid * D + dcol];
        const float ae   = 1.f / (1.f + __expf(-we * acc[r]));
        out_ae[word * D + dcol] = ae;
        wap[r] += ae * sWPW[dcol];
      }
    }

    // --- reduce word-proj dot within each 16-lane half, apply swish ---
    float sw[8];
    #pragma unroll
    for (int r = 0; r < 8; ++r) {
      float p = wap[r];
      p += __shfl_xor(p, 1, 16);
      p += __shfl_xor(p, 2, 16);
      p += __shfl_xor(p, 4, 16);
      p += __shfl_xor(p, 8, 16);
      p += wpb;
      sw[r] = p / (1.f + __expf(-p));
    }

    // --- pass 2: word_weights += sw[n] * all_embed[n,:] (ae re-read: L2-hot) ---
    for (int dt = 0; dt < 32; ++dt) {
      const int dcol = dt * 16 + mrow;
      float part = 0.f;
      #pragma unroll
      for (int r = 0; r < 8; ++r) {
        const int word = w0 + halfw + r;
        part += sw[r] * out_ae[word * D + dcol];
      }
      atomicAdd(&sWW[dcol], part);
    }
  }
  __syncthreads();
  for (int d = tid; d < D; d += 256) atomicAdd(&ww_accum[d], sWW[d]);
}

// ---------------------------------------------------------------------------
// Kernel 2: res[v] = dot(word_weights, clf_w[v,:]) + clf_b[v]  (wave per row)
// ---------------------------------------------------------------------------
__global__ __launch_bounds__(256) void clf_kernel(
    const float* __restrict__ ww,
    const float* __restrict__ clf_w,
    const float* __restrict__ clf_b,
    float* __restrict__ res)
{
  const int gw   = (blockIdx.x * 256 + threadIdx.x) >> 5;
  const int lane = threadIdx.x & 31;
  if (gw >= VOCAB) return;
  float p = 0.f;
  for (int d = lane; d < D; d += 32) p += clf_w[gw * D + d] * ww[d];
  for (int off = 16; off > 0; off >>= 1) p += __shfl_xor(p, off, 32);
  if (lane == 0) res[gw] = p + clf_b[gw];
}

// ---------------------------------------------------------------------------
extern "C" void kernel_launch(void* const* d_in, const int* in_sizes, int n_in,
                              void* d_out, int out_size, void* d_ws, size_t ws_size,
                              hipStream_t stream) {
  const int*   word_ids  = (const int*)d_in[0];
  const int*   char_ids  = (const int*)d_in[1];
  const int*   char_lens = (const int*)d_in[2];
  const float* wet       = (const float*)d_in[3];
  const float* cet       = (const float*)d_in[4];
  const float* cpw       = (const float*)d_in[5];
  const float* cpb       = (const float*)d_in[6];
  const float* wpw       = (const float*)d_in[7];
  const float* wpb       = (const float*)d_in[8];
  const float* clfw      = (const float*)d_in[9];
  const float* clfb      = (const float*)d_in[10];
  float* out = (float*)d_out;
  float* ww  = (float*)d_ws;   // 512-float word_weights accumulator

  zero_ws<<<2, 256, 0, stream>>>(ww);

  const int nblocks = (NTILES + WAVES - 1) / WAVES;  // 391
  fused_kernel<<<nblocks, 256, 0, stream>>>(word_ids, char_ids, char_lens,
                                            wet, cet, cpw, cpb, wpw, wpb,
                                            out, ww);

  clf_kernel<<<(VOCAB * 32 + 255) / 256, 256, 0, stream>>>(
      ww, clfw, clfb, out + (size_t)N_WORDS * D);
}